// HyperGraphStructuralLayer_Louvain_19825569038844
// MI455X (gfx1250) — compile-verified
//
#include <hip/hip_runtime.h>

#define NN 50000
#define HID 128
#define NF (NN * HID)
#define NF4 (NF / 4)

typedef float v2f __attribute__((ext_vector_type(2)));
typedef float v8f __attribute__((ext_vector_type(8)));

// ---------------------------------------------------------------- utilities
__global__ void k_zero4(float4* __restrict__ p, int n4) {
    int t = blockIdx.x * blockDim.x + threadIdx.x;
    if (t < n4) p[t] = make_float4(0.f, 0.f, 0.f, 0.f);
}

__global__ void k_degrees(const int* __restrict__ nidx, const int* __restrict__ eidx,
                          float* __restrict__ D, float* __restrict__ B, int E) {
    int t = blockIdx.x * blockDim.x + threadIdx.x;
    if (t < E) {
        atomicAdd(&D[nidx[t]], 1.0f);
        atomicAdd(&B[eidx[t]], 1.0f);
    }
}

__global__ void k_invert(float* __restrict__ p, int n) {
    int t = blockIdx.x * blockDim.x + threadIdx.x;
    if (t < n) {
        float v = p[t];
        p[t] = (v > 0.f) ? (1.0f / v) : 0.0f;
    }
}

// ------------------------------------------------- fp32 WMMA GEMM  Y = X @ W
// X: [nrows, 128], W: [128, 128] row-major, Y: [nrows, 128]
// block = 128 threads (4 waves); each wave computes a 16-row strip, all 8
// 16-wide N tiles, accumulating over K=128 via 32 x V_WMMA_F32_16X16X4_F32.
__global__ void __launch_bounds__(128)
k_gemm_wmma(const float* __restrict__ X, const float* __restrict__ W,
            float* __restrict__ Y, int nrows) {
    __shared__ float sW[HID * HID];          // 64 KB of the 320 KB WGP LDS
    {
        const float4* W4 = (const float4*)W;
        float4* s4 = (float4*)sW;
        for (int i = threadIdx.x; i < HID * HID / 4; i += blockDim.x)
            s4[i] = W4[i];
    }
    __syncthreads();

    const int lane = threadIdx.x & 31;
    const int wave = threadIdx.x >> 5;
    const int m0   = blockIdx.x * 64 + wave * 16;
    const int kh   = lane >> 4;              // 0: K pair {k0,k0+1}; 1: {k0+2,k0+3}
    const int nn   = lane & 15;
    int arow = m0 + nn;                      // A fragment row (M = lane%16)
    if (arow >= nrows) arow = nrows - 1;     // clamp loads; stores are guarded

    v8f c[8] = {};                           // 8 N-tiles of 16x16 f32 accum

    for (int k0 = 0; k0 < HID; k0 += 4) {
        // A frag: x[arow][k0 + 2*kh + {0,1}]  (contiguous, 8B aligned)
        v2f a = *(const v2f*)(X + (size_t)arow * HID + k0 + 2 * kh);
        const int krow = (k0 + 2 * kh) * HID + nn;
#pragma unroll
        for (int t = 0; t < 8; ++t) {
            v2f b;
            b.x = sW[krow + t * 16];         // W[k0+2*kh    ][t*16+nn]
            b.y = sW[krow + HID + t * 16];   // W[k0+2*kh + 1][t*16+nn]
            c[t] = __builtin_amdgcn_wmma_f32_16x16x4_f32(
                false, a, false, b, (short)0, c[t], false, false);
        }
    }

    // C/D layout: vgpr v -> M = v + 8*kh, N = nn (+ tile offset)
#pragma unroll
    for (int t = 0; t < 8; ++t) {
#pragma unroll
        for (int v = 0; v < 8; ++v) {
            int row = m0 + v + 8 * kh;
            if (row < nrows)
                Y[(size_t)row * HID + t * 16 + nn] = c[t][v];
        }
    }
}

// --------------------------------------------- scatter-add (segment_sum)
// dst[didx[e]][:] += src[sidx[e]][:]   -- one wave per edge, float4 per lane
__global__ void k_scatter(const float* __restrict__ src, float* __restrict__ dst,
                          const int* __restrict__ sidx, const int* __restrict__ didx,
                          int E) {
    int t = blockIdx.x * blockDim.x + threadIdx.x;
    int e = t >> 5;
    if (e >= E) return;
    int f = (t & 31) * 4;
    int s = sidx[e];
    int d = didx[e];
    float4 v = *(const float4*)(src + (size_t)s * HID + f);
    float* p = dst + (size_t)d * HID + f;
    atomicAdd(p + 0, v.x);
    atomicAdd(p + 1, v.y);
    atomicAdd(p + 2, v.z);
    atomicAdd(p + 3, v.w);
}

// m[row][:] *= Binv[row]
__global__ void k_scale_rows(float4* __restrict__ m, const float* __restrict__ inv,
                             int n4) {
    int t = blockIdx.x * blockDim.x + threadIdx.x;
    if (t >= n4) return;
    float s = inv[t >> 5];                   // 32 float4 per 128-wide row
    float4 v = m[t];
    v.x *= s; v.y *= s; v.z *= s; v.w *= s;
    m[t] = v;
}

__device__ __forceinline__ float prelu(float v, float a) {
    return v >= 0.f ? v : a * v;
}

// h = prelu(agg * Dinv[row] + bias)
__global__ void k_finalize1(const float4* __restrict__ agg, const float* __restrict__ Dinv,
                            const float4* __restrict__ bias, const float* __restrict__ pa,
                            float4* __restrict__ out, int n4) {
    int t = blockIdx.x * blockDim.x + threadIdx.x;
    if (t >= n4) return;
    float s = Dinv[t >> 5];
    float a = pa[0];
    float4 v = agg[t];
    float4 b = bias[t & 31];
    v.x = prelu(v.x * s + b.x, a);
    v.y = prelu(v.y * s + b.y, a);
    v.z = prelu(v.z * s + b.z, a);
    v.w = prelu(v.w * s + b.w, a);
    out[t] = v;
}

// out = prelu(agg * Dinv[row] + bias + x_residual)
__global__ void k_finalize2(const float4* __restrict__ agg, const float* __restrict__ Dinv,
                            const float4* __restrict__ bias, const float* __restrict__ pa,
                            const float4* __restrict__ xres, float4* __restrict__ out,
                            int n4) {
    int t = blockIdx.x * blockDim.x + threadIdx.x;
    if (t >= n4) return;
    float s = Dinv[t >> 5];
    float a = pa[0];
    float4 v = agg[t];
    float4 b = bias[t & 31];
    float4 r = xres[t];
    v.x = prelu(v.x * s + b.x + r.x, a);
    v.y = prelu(v.y * s + b.y + r.y, a);
    v.z = prelu(v.z * s + b.z + r.z, a);
    v.w = prelu(v.w * s + b.w + r.w, a);
    out[t] = v;
}

// ---------------------------------------------------------------- launcher
extern "C" void kernel_launch(void* const* d_in, const int* in_sizes, int n_in,
                              void* d_out, int out_size, void* d_ws, size_t ws_size,
                              hipStream_t stream) {
    (void)n_in; (void)out_size; (void)ws_size;

    const float* x    = (const float*)d_in[0];
    const int*   hei  = (const int*)d_in[1];
    const float* W1   = (const float*)d_in[2];
    const float* b1   = (const float*)d_in[3];
    const float* W2   = (const float*)d_in[4];
    const float* b2   = (const float*)d_in[5];
    const float* pa   = (const float*)d_in[6];
    float*       out  = (float*)d_out;

    const int E = in_sizes[1] / 2;
    const int* nidx = hei;       // hyper_edge_index[0]
    const int* eidx = hei + E;   // hyper_edge_index[1]

    float* ws   = (float*)d_ws;
    float* buf0 = ws;                 // xw / agg
    float* buf1 = ws + (size_t)NF;    // m (hyperedge features)
    float* buf2 = ws + (size_t)2 * NF;// h (layer-1 output)
    float* Ddeg = ws + (size_t)3 * NF;
    float* Bdeg = Ddeg + NN;          // D and B contiguous (2*NN floats)

    const int TB = 256;
    const int gElem   = (NF4 + TB - 1) / TB;            // float4 elementwise grids
    const int gDeg    = (2 * NN / 4 + TB - 1) / TB;     // zero D+B (float4)
    const int gEdges  = (E + TB - 1) / TB;
    const int gScat   = (E * 32 + TB - 1) / TB;
    const int gInv    = (2 * NN + TB - 1) / TB;
    const int gGemm   = (NN + 63) / 64;

    // degrees (shared by both layers)
    k_zero4<<<gDeg, TB, 0, stream>>>((float4*)Ddeg, 2 * NN / 4);
    k_degrees<<<gEdges, TB, 0, stream>>>(nidx, eidx, Ddeg, Bdeg, E);
    k_invert<<<gInv, TB, 0, stream>>>(Ddeg, 2 * NN);    // D,B -> Dinv,Binv in place

    // ---- layer 1 ----
    k_gemm_wmma<<<gGemm, 128, 0, stream>>>(x, W1, buf0, NN);
    k_zero4<<<gElem, TB, 0, stream>>>((float4*)buf1, NF4);
    k_scatter<<<gScat, TB, 0, stream>>>(buf0, buf1, nidx, eidx, E);   // node -> hedge
    k_scale_rows<<<gElem, TB, 0, stream>>>((float4*)buf1, Bdeg, NF4); // * B^-1
    k_zero4<<<gElem, TB, 0, stream>>>((float4*)buf0, NF4);
    k_scatter<<<gScat, TB, 0, stream>>>(buf1, buf0, eidx, nidx, E);   // hedge -> node
    k_finalize1<<<gElem, TB, 0, stream>>>((const float4*)buf0, Ddeg,
                                          (const float4*)b1, pa,
                                          (float4*)buf2, NF4);        // h = prelu(.)

    // ---- layer 2 ----
    k_gemm_wmma<<<gGemm, 128, 0, stream>>>(buf2, W2, buf0, NN);
    k_zero4<<<gElem, TB, 0, stream>>>((float4*)buf1, NF4);
    k_scatter<<<gScat, TB, 0, stream>>>(buf0, buf1, nidx, eidx, E);
    k_scale_rows<<<gElem, TB, 0, stream>>>((float4*)buf1, Bdeg, NF4);
    k_zero4<<<gElem, TB, 0, stream>>>((float4*)buf0, NF4);
    k_scatter<<<gScat, TB, 0, stream>>>(buf1, buf0, eidx, nidx, E);
    k_finalize2<<<gElem, TB, 0, stream>>>((const float4*)buf0, Ddeg,
                                          (const float4*)b2, pa,
                                          (const float4*)x, (float4*)out, NF4);
}